// Edge_to_Node_Conv2d_79164837200688
// MI455X (gfx1250) — compile-verified
//
#include <hip/hip_runtime.h>

typedef float v2f __attribute__((ext_vector_type(2)));
typedef float v8f __attribute__((ext_vector_type(8)));

#define KDIM 1024
#define NBATCH 128
#define ROWS_PER_BLOCK 64
#define RGROUPS 4            // 4 row-groups of 16 rows
#define JTILE 256            // column super-tile width
#define NJT (KDIM / JTILE)   // 4
#define TPB 128              // 4 waves (wave32)
#define PITCH 260            // 256 + 4 pad: 260 mod 64 = 4 -> conflict-free strided A reads
#define NROWBLK (KDIM / ROWS_PER_BLOCK) // 16 row-blocks per batch

// Pass 1: fused col-GEMV + row-partial-GEMV over one streaming read of x.
// grid = (NROWBLK, NBATCH), block = 128 threads.
__global__ __launch_bounds__(TPB) void fused_colrow_kernel(
    const float* __restrict__ x, const float* __restrict__ w_col,
    const float* __restrict__ b_col_p, const float* __restrict__ w_row,
    float* __restrict__ col_out, float* __restrict__ row_part)
{
    __shared__ float tile[16 * PITCH];        // 16 x 256 fp32 X tile (padded)
    __shared__ float wcol_s[KDIM];
    __shared__ float wrow_s[ROWS_PER_BLOCK];
    __shared__ float row_acc[KDIM];           // per-block row partials
    __shared__ float col_sums[4 * 16];        // per-wave col partial C extraction

    const int tid  = threadIdx.x;
    const int wave = tid >> 5;
    const int lane = tid & 31;
    const int m    = lane & 15;
    const int hi   = lane >> 4;               // 0: lanes 0-15, 1: lanes 16-31
    const int b    = blockIdx.y;
    const int rb   = blockIdx.x;              // row-block 0..15
    const int i0base = rb * ROWS_PER_BLOCK;

    const float* xb = x + (size_t)b * KDIM * KDIM;

    for (int j = tid; j < KDIM; j += TPB) { wcol_s[j] = w_col[j]; row_acc[j] = 0.0f; }
    if (tid < ROWS_PER_BLOCK) wrow_s[tid] = w_row[i0base + tid];
    __syncthreads();

    for (int rg = 0; rg < RGROUPS; ++rg) {
        const int i0 = i0base + rg * 16;
        v8f c_col = {};                       // 16x16 f32 accum (col GEMV, all N equal)

        for (int jt = 0; jt < NJT; ++jt) {
            __syncthreads();                  // previous tile fully consumed
            // ---- cooperative coalesced load: 16 rows x 256 cols (b128) ----
            for (int q = 0; q < 8; ++q) {
                int idx = q * TPB + tid;      // 0..1023 float4 slots
                int r   = idx >> 6;           // 64 float4 per row
                int c4  = idx & 63;
                float4 v = *(const float4*)(xb + (size_t)(i0 + r) * KDIM
                                               + jt * JTILE + (c4 << 2));
                *(float4*)(&tile[r * PITCH + (c4 << 2)]) = v;
            }
            __syncthreads();
            if (jt + 1 < NJT)                 // pull next tile toward L2/L0
                __builtin_prefetch(xb + (size_t)i0 * KDIM + (jt + 1) * JTILE, 0, 1);

            // ---- col GEMV: C += X[16x4] * wcol_bcast[4x16], k split across waves ----
            {
                const int kbase = wave * 64;
                for (int kk = kbase; kk < kbase + 64; kk += 4) {
                    // A 16x4 layout: lanes0-15 K=0(+1), lanes16-31 K=2(+3)
                    v2f a  = *(const v2f*)(&tile[m * PITCH + kk + 2 * hi]);
                    // B 4x16 (replicated over N): VGPR0 K=0/2, VGPR1 K=1/3
                    v2f bv = *(const v2f*)(&wcol_s[jt * JTILE + kk + 2 * hi]);
                    c_col = __builtin_amdgcn_wmma_f32_16x16x4_f32(
                        false, a, false, bv, (short)0, c_col, false, false);
                }
            }

            // ---- row partials: C[m,n] = sum_i w_row[i]*X[i, jj+n], 4 sub-tiles/wave ----
            for (int s = 0; s < 4; ++s) {
                const int jj = wave * 64 + s * 16;
                v8f c_row = {};
                for (int ii = 0; ii < 16; ii += 4) {
                    // A 16x4 replicated over M: A[m,k] = w_row[i0 + ii + k]
                    v2f a = *(const v2f*)(&wrow_s[rg * 16 + ii + 2 * hi]);
                    // B[k,n] = X[ii+k, jj+n]
                    v2f bv;
                    bv.x = tile[(ii +     2 * hi) * PITCH + jj + m];
                    bv.y = tile[(ii + 1 + 2 * hi) * PITCH + jj + m];
                    c_row = __builtin_amdgcn_wmma_f32_16x16x4_f32(
                        false, a, false, bv, (short)0, c_row, false, false);
                }
                // every C row is identical; lanes 0-15 of VGPR0 hold N=0..15 (M=0)
                if (hi == 0) row_acc[jt * JTILE + jj + m] += c_row[0];
            }
        }

        // ---- extract col for this 16-row group: C[:,0] lives in lanes 0 and 16 ----
        if (lane == 0) {
            for (int v = 0; v < 8; ++v) col_sums[wave * 16 + v] = c_col[v];
        } else if (lane == 16) {
            for (int v = 0; v < 8; ++v) col_sums[wave * 16 + 8 + v] = c_col[v];
        }
        __syncthreads();
        if (tid < 16) {
            float cv = col_sums[tid] + col_sums[16 + tid] +
                       col_sums[32 + tid] + col_sums[48 + tid] + b_col_p[0];
            col_out[(size_t)b * KDIM + i0 + tid] = cv;
        }
        __syncthreads();                      // col_sums reused next rg
    }

    // deterministic per-block row partials (no float atomics)
    for (int j = tid; j < KDIM; j += TPB)
        row_part[((size_t)b * NROWBLK + rb) * KDIM + j] = row_acc[j];
}

// Pass 2: reduce row partials, S[b] = col . row, broadcast to output row.
__global__ __launch_bounds__(256) void finalize_kernel(
    const float* __restrict__ col_out, const float* __restrict__ row_part,
    const float* __restrict__ b_row_p, float* __restrict__ out)
{
    __shared__ float red[256];
    const int b   = blockIdx.x;
    const int tid = threadIdx.x;
    const float b_row = b_row_p[0];

    float acc = 0.0f;
    for (int j = tid; j < KDIM; j += 256) {
        float r = b_row;
        for (int c = 0; c < NROWBLK; ++c)
            r += row_part[((size_t)b * NROWBLK + c) * KDIM + j];
        acc += col_out[(size_t)b * KDIM + j] * r;
    }
    red[tid] = acc;
    __syncthreads();
    for (int s = 128; s > 0; s >>= 1) {
        if (tid < s) red[tid] += red[tid + s];
        __syncthreads();
    }
    const float S = red[0];
    for (int j = tid; j < KDIM; j += 256)
        out[(size_t)b * KDIM + j] = S;        // out shape (B,1,K) flattened
}

extern "C" void kernel_launch(void* const* d_in, const int* in_sizes, int n_in,
                              void* d_out, int out_size, void* d_ws, size_t ws_size,
                              hipStream_t stream) {
    const float* x     = (const float*)d_in[0];   // (128,1,1024,1024)
    const float* w_col = (const float*)d_in[1];   // (1024)
    const float* b_col = (const float*)d_in[2];   // scalar
    const float* w_row = (const float*)d_in[3];   // (1024)
    const float* b_row = (const float*)d_in[4];   // scalar
    float* out = (float*)d_out;                   // (128,1,1024)

    float* col_out  = (float*)d_ws;                               // 128*1024 f32
    float* row_part = col_out + (size_t)NBATCH * KDIM;            // 128*16*1024 f32

    fused_colrow_kernel<<<dim3(NROWBLK, NBATCH), TPB, 0, stream>>>(
        x, w_col, b_col, w_row, col_out, row_part);
    finalize_kernel<<<NBATCH, 256, 0, stream>>>(col_out, row_part, b_row, out);
}